// Dynamic_relu_b_31044023616421
// MI455X (gfx1250) — compile-verified
//
#include <hip/hip_runtime.h>
#include <hip/hip_bf16.h>

// ---------------------------------------------------------------------------
// Dynamic ReLU-B on MI455X (gfx1250, wave32)
//   Pass 1: per-(b,c) mean pool (x stays resident in 192MB L2)
//   Pass 2: tiny FC stack done with V_WMMA_F32_16X16X4_F32 (full fp32 WMMA)
//   Pass 3: streaming apply, max over K=2, non-temporal stores for the output
// ---------------------------------------------------------------------------

typedef __attribute__((ext_vector_type(2))) float v2f;
typedef __attribute__((ext_vector_type(4))) float f4v;
typedef __attribute__((ext_vector_type(8))) float v8f;

#define B_DIM   32
#define C_DIM   256
#define HW      3136          // 56*56
#define HW4     784           // HW/4
#define HID     32            // C/R, R=8
#define KSEL    2
#define O_DIM   512           // 2*C

// ---------------------------------------------------------------------------
// Kernel 1: pooled[b*C+c] = mean(x[b,c,:,:]).  One block per (b,c).
// ---------------------------------------------------------------------------
__global__ __launch_bounds__(256) void pool_kernel(const float* __restrict__ x,
                                                   float* __restrict__ pooled) {
    const int bc = blockIdx.x;                       // 0 .. B*C-1
    const f4v* xv = (const f4v*)(x + (size_t)bc * HW);

    float s = 0.0f;
    for (int i = threadIdx.x; i < HW4; i += 256) {
        f4v v = xv[i];                               // regular-temporal: keep x in L2
        s += v.x + v.y + v.z + v.w;
    }

    __shared__ float red[256];
    red[threadIdx.x] = s;
    __syncthreads();
    #pragma unroll
    for (int off = 128; off > 0; off >>= 1) {
        if (threadIdx.x < off) red[threadIdx.x] += red[threadIdx.x + off];
        __syncthreads();
    }
    if (threadIdx.x == 0) pooled[bc] = red[0] * (1.0f / (float)HW);
}

// ---------------------------------------------------------------------------
// Kernel 2: FC1 -> ReLU -> FC2 -> 2*sigmoid-1 -> a/b coefficient planes.
// Single workgroup, 4 wave32s, all matrix math on V_WMMA_F32_16X16X4_F32.
//
// fp32 WMMA operand layout (ISA 7.12.2):
//   A (16x4):  lane L holds row M=L&15; lanes 0-15 -> K = k0+{0,1} in the two
//              VGPRs, lanes 16-31 -> K = k0+{2,3}.
//   B (4x16):  lane L holds col N=L&15; same K split across lane halves.
//   C/D (16x16): VGPR v, lanes 0-15 -> row v, lanes 16-31 -> row v+8; N = lane&15.
// ---------------------------------------------------------------------------
__global__ __launch_bounds__(128) void fc_kernel(const float* __restrict__ pooled,
                                                 const float* __restrict__ fc1_w,
                                                 const float* __restrict__ fc1_b,
                                                 const float* __restrict__ fc2_w,
                                                 const float* __restrict__ fc2_b,
                                                 float* __restrict__ a_coef,
                                                 float* __restrict__ b_coef) {
    const int tid  = threadIdx.x;
    const int wave = tid >> 5;
    const int lane = tid & 31;
    const int lr   = lane & 15;
    const int kOff = (lane < 16) ? 0 : 2;       // K sub-offset for A/B operands
    const int hiHf = (lane < 16) ? 0 : 8;       // row offset for C/D readout

    __shared__ float hact[B_DIM * HID];         // 32x32 activations, 4 KB LDS

    // ---------------- GEMM1: h = relu(pooled(32x256) @ fc1_w^T + fc1_b) ----
    {
        const int mt = wave >> 1;               // 2x2 grid of 16x16 tiles
        const int nt = wave & 1;
        const int aRow = mt * 16 + lr;          // batch index
        const int bCol = nt * 16 + lr;          // hidden index (B column)
        v8f acc = {};
        for (int k0 = 0; k0 < C_DIM; k0 += 4) {
            v2f a, b;
            a.x = pooled[aRow * C_DIM + k0 + kOff];
            a.y = pooled[aRow * C_DIM + k0 + kOff + 1];
            // B[k][n] = fc1_w[n][k]
            b.x = fc1_w[bCol * C_DIM + k0 + kOff];
            b.y = fc1_w[bCol * C_DIM + k0 + kOff + 1];
            acc = __builtin_amdgcn_wmma_f32_16x16x4_f32(
                    false, a, false, b, (short)0, acc, false, false);
        }
        const float bias = fc1_b[bCol];
        #pragma unroll
        for (int v = 0; v < 8; ++v) {
            const int m = mt * 16 + v + hiHf;
            const float h = acc[v] + bias;
            hact[m * HID + bCol] = h > 0.0f ? h : 0.0f;
        }
    }
    __syncthreads();

    // ------ GEMM2: delta[k,b,o] = sum_h fc2_w[k,o,h] * hact[b,h] + fc2_b ---
    // 128 tiles of 16x16: k in {0,1} x mt2 in {0,1} x nt2 in 0..31
    for (int t = wave; t < 128; t += 4) {
        const int k   = t >> 6;
        const int mt2 = (t >> 5) & 1;
        const int nt2 = t & 31;
        const int aRow = mt2 * 16 + lr;         // batch index
        const int o    = nt2 * 16 + lr;         // output channel 0..511
        const float* w = fc2_w + (size_t)k * O_DIM * HID;
        v8f acc = {};
        #pragma unroll
        for (int k0 = 0; k0 < HID; k0 += 4) {
            v2f a, b;
            a.x = hact[aRow * HID + k0 + kOff];
            a.y = hact[aRow * HID + k0 + kOff + 1];
            // B[h][o] = fc2_w[k][o][h]
            b.x = w[o * HID + k0 + kOff];
            b.y = w[o * HID + k0 + kOff + 1];
            acc = __builtin_amdgcn_wmma_f32_16x16x4_f32(
                    false, a, false, b, (short)0, acc, false, false);
        }
        const float bias  = fc2_b[k * O_DIM + o];
        const float initv = (k == 0) ? 1.0f : 0.0f;
        const int   c     = o >> 1;
        const bool  isAlpha = (o & 1) == 0;
        #pragma unroll
        for (int v = 0; v < 8; ++v) {
            const int m = mt2 * 16 + v + hiHf;  // batch index
            float d = acc[v] + bias;
            d = 2.0f / (1.0f + __expf(-d)) - 1.0f;       // 2*sigmoid(d)-1
            if (isAlpha)   // LAMBDA_ALPHA = 1.0
                a_coef[((size_t)k * B_DIM + m) * C_DIM + c] = initv + d;
            else           // LAMBDA_BETA = 0.5
                b_coef[((size_t)k * B_DIM + m) * C_DIM + c] = initv + 0.5f * d;
        }
    }
}

// ---------------------------------------------------------------------------
// Kernel 3: out[b,c,:] = max_k( x[b,c,:] * a[k,b,c] + b[k,b,c] )
// One block per (b,c); x re-read from L2, output stored non-temporally so the
// streaming write does not evict x / coefficients from the 192 MB L2.
// ---------------------------------------------------------------------------
__global__ __launch_bounds__(256) void apply_kernel(const float* __restrict__ x,
                                                    const float* __restrict__ a_coef,
                                                    const float* __restrict__ b_coef,
                                                    float* __restrict__ out) {
    const int bc = blockIdx.x;                  // b*C + c
    const float a0 = a_coef[bc];
    const float b0 = b_coef[bc];
    const float a1 = a_coef[B_DIM * C_DIM + bc];
    const float b1 = b_coef[B_DIM * C_DIM + bc];

    const f4v* xv = (const f4v*)(x + (size_t)bc * HW);
    f4v*       ov = (f4v*)(out + (size_t)bc * HW);

    for (int i = threadIdx.x; i < HW4; i += 256) {
        f4v v = xv[i];
        f4v r;
        r.x = fmaxf(fmaf(v.x, a0, b0), fmaf(v.x, a1, b1));
        r.y = fmaxf(fmaf(v.y, a0, b0), fmaf(v.y, a1, b1));
        r.z = fmaxf(fmaf(v.z, a0, b0), fmaf(v.z, a1, b1));
        r.w = fmaxf(fmaf(v.w, a0, b0), fmaf(v.w, a1, b1));
        __builtin_nontemporal_store(r, ov + i);
    }
}

// ---------------------------------------------------------------------------
extern "C" void kernel_launch(void* const* d_in, const int* in_sizes, int n_in,
                              void* d_out, int out_size, void* d_ws, size_t ws_size,
                              hipStream_t stream) {
    const float* x     = (const float*)d_in[0];
    const float* fc1_w = (const float*)d_in[1];
    const float* fc1_b = (const float*)d_in[2];
    const float* fc2_w = (const float*)d_in[3];
    const float* fc2_b = (const float*)d_in[4];
    float*       out   = (float*)d_out;

    // workspace layout (floats): pooled[8192] | a_coef[2*8192] | b_coef[2*8192]
    float* ws     = (float*)d_ws;
    float* pooled = ws;
    float* a_coef = ws + B_DIM * C_DIM;
    float* b_coef = a_coef + KSEL * B_DIM * C_DIM;

    pool_kernel <<<B_DIM * C_DIM, 256, 0, stream>>>(x, pooled);
    fc_kernel   <<<1, 128, 0, stream>>>(pooled, fc1_w, fc1_b, fc2_w, fc2_b,
                                        a_coef, b_coef);
    apply_kernel<<<B_DIM * C_DIM, 256, 0, stream>>>(x, a_coef, b_coef, out);
}